// a_sum_op_6631429505523
// MI455X (gfx1250) — compile-verified
//
#include <hip/hip_runtime.h>
#include <stdint.h>

// Problem constants (from the reference).
#define ND 100000      // N_DST
#define NE 1600000     // N_EDGES
#define DD 128         // D

typedef float v4f __attribute__((ext_vector_type(4)));  // clang vector: OK for NT builtins

// ---------------------------------------------------------------------------
// Kernel 1: zero the accumulator (d_out doubles as the segment-sum buffer).
// Regular-temporal stores so the zeroed lines are L2-resident when the
// atomics start hammering them.
// ---------------------------------------------------------------------------
__global__ __launch_bounds__(256) void zero_out_kernel(v4f* __restrict__ out, int n4) {
  int i = blockIdx.x * blockDim.x + threadIdx.x;
  if (i < n4) out[i] = (v4f){0.f, 0.f, 0.f, 0.f};
}

// ---------------------------------------------------------------------------
// Kernel 2: scatter-add with CDNA5 async global->LDS staging.
// One wave owns one 8-edge tile at a time: 8 x (32 lanes x 16B) = 4 KB moved
// per tile with 8 global_load_async_to_lds_b128 instructions (ASYNCcnt),
// double-buffered so the next tile's DMA overlaps the current tile's atomics.
// The message stream (819 MB, read once) is loaded non-temporal so it does
// not evict the L2-resident 51 MB accumulator; atomics absorb in L2.
// ---------------------------------------------------------------------------
#define EPT 8   // edges per wave-tile
#define WPB 8   // waves per 256-thread block

__global__ __launch_bounds__(256) void scatter_kernel(
    const float* __restrict__ msgs, const int* __restrict__ dst_idx,
    float* __restrict__ out, int n_tiles, int total_waves)
{
  __shared__ float lds[2][WPB][EPT][DD];  // 2 * 8 * 8 * 128 * 4B = 64 KB
  const int lane = threadIdx.x & 31;
  const int wave = threadIdx.x >> 5;
  int t = blockIdx.x * WPB + wave;        // wave-uniform
  if (t >= n_tiles) return;

  const float* gsrc_lane = msgs + (size_t)lane * 4;

  // Issue one tile (8 edge rows) of async copies into LDS buffer b.
  auto issue = [&](int tile, int b) {
    const float* g = gsrc_lane + (size_t)tile * EPT * DD;
    // Low 32 bits of a generic pointer to a __shared__ object are the LDS
    // byte offset (aperture lives in the high bits) -> usable as VDST addr.
    uint32_t la = (uint32_t)(uintptr_t)&lds[b][wave][0][lane * 4];
#pragma unroll
    for (int k = 0; k < EPT; ++k) {
      uint64_t ga = (uint64_t)(uintptr_t)(g + (size_t)k * DD);
      uint32_t lk = la + k * DD * 4;
      asm volatile("global_load_async_to_lds_b128 %0, %1, off th:TH_LOAD_NT"
                   :: "v"(lk), "v"(ga) : "memory");
    }
  };

  // Consume one staged tile: ds_load_b128 per lane, 4 f32 atomics per lane.
  auto consume = [&](int tile, int b) {
#pragma unroll
    for (int k = 0; k < EPT; ++k) {
      const int e = tile * EPT + k;
      const int dst = dst_idx[e];  // wave-uniform -> broadcast load
      const v4f m = *(const v4f*)&lds[b][wave][k][lane * 4];
      float* o = out + (size_t)dst * DD + lane * 4;
      unsafeAtomicAdd(o + 0, m.x);
      unsafeAtomicAdd(o + 1, m.y);
      unsafeAtomicAdd(o + 2, m.z);
      unsafeAtomicAdd(o + 3, m.w);
    }
  };

  int b = 0;
  issue(t, b);
  for (;;) {
    const int t2 = t + total_waves;
    if (t2 < n_tiles) {
      issue(t2, b ^ 1);                                // 8 more in flight (cnt<=16)
      asm volatile("s_wait_asynccnt 8" ::: "memory");  // first 8 (tile t) done
      consume(t, b);
      t = t2;
      b ^= 1;
    } else {
      asm volatile("s_wait_asynccnt 0" ::: "memory");
      consume(t, b);
      break;
    }
  }
}

// ---------------------------------------------------------------------------
// Kernel 3: dropout (bit-exact JAX threefry2x32, key = (0, 42)) + bias add.
// JAX pairs flat element j with j+half in one threefry call. Each thread
// finalizes 4 consecutive elements of BOTH halves (4 threefry calls), so all
// global traffic is b128: 2 loads (h), 2 NT loads (bias), 2 NT stores (out).
// ---------------------------------------------------------------------------
__device__ __forceinline__ uint32_t rotl32(uint32_t x, uint32_t r) {
  return (x << r) | (x >> (32u - r));
}

// Returns (out0, out1) of threefry2x32 with key (0, 42) for counters (c0, c1).
__device__ __forceinline__ void threefry2x32_0_42(uint32_t c0, uint32_t c1,
                                                  uint32_t& o0, uint32_t& o1) {
  const uint32_t k0 = 0u, k1 = 42u;
  const uint32_t k2 = 0x1BD11BDAu ^ k0 ^ k1;
  uint32_t x0 = c0 + k0;
  uint32_t x1 = c1 + k1;
  auto quad = [&](uint32_t ra, uint32_t rb, uint32_t rc, uint32_t rd) {
    x0 += x1; x1 = rotl32(x1, ra); x1 ^= x0;
    x0 += x1; x1 = rotl32(x1, rb); x1 ^= x0;
    x0 += x1; x1 = rotl32(x1, rc); x1 ^= x0;
    x0 += x1; x1 = rotl32(x1, rd); x1 ^= x0;
  };
  quad(13, 15, 26, 6);   x0 += k1; x1 += k2 + 1u;
  quad(17, 29, 16, 24);  x0 += k2; x1 += k0 + 2u;
  quad(13, 15, 26, 6);   x0 += k0; x1 += k1 + 3u;
  quad(17, 29, 16, 24);  x0 += k1; x1 += k2 + 4u;
  quad(13, 15, 26, 6);   x0 += k2; x1 += k0 + 5u;
  o0 = x0; o1 = x1;
}

__device__ __forceinline__ float tf_uniform(uint32_t bits) {
  return __uint_as_float((bits >> 9) | 0x3f800000u) - 1.0f;
}

__global__ __launch_bounds__(256) void finalize_kernel(
    const float* __restrict__ bias, float* __restrict__ out, int half)
{
  const int i4 = blockIdx.x * blockDim.x + threadIdx.x;  // quad index
  const int base = i4 * 4;
  if (base >= half) return;

  constexpr float KEEP = 0.9f;
  constexpr float SCL  = 1.0f / 0.9f;

  v4f h0 = *(const v4f*)(out + base);
  v4f h1 = *(const v4f*)(out + half + base);
  v4f b0 = __builtin_nontemporal_load((const v4f*)(bias + base));
  v4f b1 = __builtin_nontemporal_load((const v4f*)(bias + half + base));

  v4f r0, r1;
#pragma unroll
  for (int c = 0; c < 4; ++c) {
    const uint32_t j = (uint32_t)(base + c);
    uint32_t bits0, bits1;
    threefry2x32_0_42(j, j + (uint32_t)half, bits0, bits1);
    const float u0 = tf_uniform(bits0);
    const float u1 = tf_uniform(bits1);
    r0[c] = (u0 < KEEP ? h0[c] * SCL : 0.0f) + b0[c];
    r1[c] = (u1 < KEEP ? h1[c] * SCL : 0.0f) + b1[c];
  }

  __builtin_nontemporal_store(r0, (v4f*)(out + base));
  __builtin_nontemporal_store(r1, (v4f*)(out + half + base));
}

// ---------------------------------------------------------------------------
// Launcher: zero -> async-staged scatter-add -> threefry dropout + bias.
// Inputs: d_in[0]=src_emb (NE+ND, D) f32, d_in[1]=src_emb_in (unused by the
// reference output), d_in[2]=dst_idx (NE,) i32. Output: (ND, D) f32.
// ---------------------------------------------------------------------------
extern "C" void kernel_launch(void* const* d_in, const int* in_sizes, int n_in,
                              void* d_out, int out_size, void* d_ws, size_t ws_size,
                              hipStream_t stream) {
  (void)in_sizes; (void)n_in; (void)out_size; (void)d_ws; (void)ws_size;
  const float* src_emb = (const float*)d_in[0];
  const int*   dst_idx = (const int*)d_in[2];
  float*       out     = (float*)d_out;

  const float* msgs = src_emb;                         // rows [0, NE)
  const float* bias = src_emb + (size_t)NE * DD;       // rows [NE, NE+ND)

  // 1) zero accumulator (12.8M floats as float4)
  const int n4 = ND * DD / 4;  // 3,200,000
  zero_out_kernel<<<(n4 + 255) / 256, 256, 0, stream>>>((v4f*)out, n4);

  // 2) scatter-add: 200,000 wave-tiles; 3125 blocks * 8 waves = 25,000 waves,
  //    8 tiles per wave -> double buffering actually overlaps.
  const int n_tiles = NE / EPT;        // 200,000
  const int blocks  = 3125;
  const int total_waves = blocks * WPB;
  scatter_kernel<<<blocks, 256, 0, stream>>>(msgs, dst_idx, out, n_tiles, total_waves);

  // 3) dropout + bias: each thread finalizes 4 elems of both halves (b128).
  const int half = ND * DD / 2;        // 6,400,000
  const int quads = half / 4;          // 1,600,000
  finalize_kernel<<<(quads + 255) / 256, 256, 0, stream>>>(bias, out, half);
}